// GENConv_79697413144781
// MI455X (gfx1250) — compile-verified
//
#include <hip/hip_runtime.h>

#define DF 96            // feature dim (in == out)
#define NTILES 6         // DF / 16
#define TILE_ROWS 64     // rows of feats per block in GEMM kernel
#define GEMM_THREADS 128 // 4 waves, each owning 16 rows
#define FEPS 1e-7f

typedef __attribute__((ext_vector_type(2))) float v2f;
typedef __attribute__((ext_vector_type(8))) float v8f;

// ---------------------------------------------------------------------------
// Workspace layout (d_ws):
//   [0            , N*DF)  : unsigned int  per-(node,dim) running max (float bits)
//   [N*DF         , 2N*DF) : float         numerator  sum(exp(m-max)*m)
//   [2N*DF        , 3N*DF) : float         denominator sum(exp(m-max))
// ---------------------------------------------------------------------------

__global__ __launch_bounds__(256) void genconv_zero_ws(unsigned int* __restrict__ p,
                                                       int n) {
  int i = blockIdx.x * 256 + threadIdx.x;
  if (i < n) p[i] = 0u;   // 0u == +0.0f bits; valid identity since m > 0
}

// Pass 1: per-(dst, feature) max of m = relu(feat[src]) + eps.
// m > 0 always, so unsigned-int compare on float bits == float compare.
__global__ __launch_bounds__(256) void genconv_edge_max(
    const float* __restrict__ feat, const int* __restrict__ src,
    const int* __restrict__ dst, unsigned int* __restrict__ mx, int E) {
  int idx = blockIdx.x * 256 + threadIdx.x;
  if (idx >= E * 24) return;          // 24 float4 quads per 96-wide row
  int e = idx / 24;
  int q = idx % 24;
  int s = src[e];
  int d = dst[e];
  const float4* fr = reinterpret_cast<const float4*>(feat + (size_t)s * DF);
  float4 f = fr[q];
  unsigned int* mrow = mx + (size_t)d * DF + q * 4;
  atomicMax(mrow + 0, __float_as_uint(fmaxf(f.x, 0.f) + FEPS));
  atomicMax(mrow + 1, __float_as_uint(fmaxf(f.y, 0.f) + FEPS));
  atomicMax(mrow + 2, __float_as_uint(fmaxf(f.z, 0.f) + FEPS));
  atomicMax(mrow + 3, __float_as_uint(fmaxf(f.w, 0.f) + FEPS));
}

// Pass 2: fused softmax numerator + denominator:
//   num += exp(m - max) * m ;  den += exp(m - max)
__global__ __launch_bounds__(256) void genconv_edge_sum(
    const float* __restrict__ feat, const int* __restrict__ src,
    const int* __restrict__ dst, const unsigned int* __restrict__ mx,
    float* __restrict__ num, float* __restrict__ den, int E) {
  int idx = blockIdx.x * 256 + threadIdx.x;
  if (idx >= E * 24) return;
  int e = idx / 24;
  int q = idx % 24;
  int s = src[e];
  int d = dst[e];
  const float4* fr = reinterpret_cast<const float4*>(feat + (size_t)s * DF);
  float4 f = fr[q];
  size_t base = (size_t)d * DF + q * 4;
  const unsigned int* mrow = mx + base;
  float m0 = fmaxf(f.x, 0.f) + FEPS;
  float m1 = fmaxf(f.y, 0.f) + FEPS;
  float m2 = fmaxf(f.z, 0.f) + FEPS;
  float m3 = fmaxf(f.w, 0.f) + FEPS;
  float e0 = __expf(m0 - __uint_as_float(mrow[0]));
  float e1 = __expf(m1 - __uint_as_float(mrow[1]));
  float e2 = __expf(m2 - __uint_as_float(mrow[2]));
  float e3 = __expf(m3 - __uint_as_float(mrow[3]));
  atomicAdd(num + base + 0, e0 * m0);
  atomicAdd(num + base + 1, e1 * m1);
  atomicAdd(num + base + 2, e2 * m2);
  atomicAdd(num + base + 3, e3 * m3);
  atomicAdd(den + base + 0, e0);
  atomicAdd(den + base + 1, e1);
  atomicAdd(den + base + 2, e2);
  atomicAdd(den + base + 3, e3);
}

// Fused: feats = feat + num/den (0 if no incoming edges), out = feats @ W + b.
// One block = 4 wave32s, 64 rows. W (96x96, 36KB) + feats tile (64x96, 24KB)
// staged in LDS; each wave computes a 16x96 strip with V_WMMA_F32_16X16X4_F32.
__global__ __launch_bounds__(GEMM_THREADS) void genconv_fused_gemm(
    const float* __restrict__ feat, const float* __restrict__ num,
    const float* __restrict__ den, const float* __restrict__ W,
    const float* __restrict__ b, float* __restrict__ out, int N) {
  __shared__ float sW[DF * DF];          // 36 KB, row-major [k][n]
  __shared__ float sA[TILE_ROWS * DF];   // 24 KB, row-major [m][k]

  const int tid = threadIdx.x;
  const int rowBase = blockIdx.x * TILE_ROWS;

  for (int i = tid; i < DF * DF; i += GEMM_THREADS) sW[i] = W[i];

  for (int i = tid; i < TILE_ROWS * DF; i += GEMM_THREADS) {
    int r = rowBase + i / DF;
    float v = 0.f;
    if (r < N) {
      size_t gi = (size_t)r * DF + (i % DF);
      float dn = den[gi];
      float msg = (dn > 0.f) ? (num[gi] / dn) : 0.f;  // isolated node -> 0
      v = feat[gi] + msg;                              // residual
    }
    sA[i] = v;
  }
  __syncthreads();

  const int wave = tid >> 5;
  const int lane = tid & 31;
  const int mrow = lane & 15;  // M (for A) / N (for B,C) within tile
  const int kh   = lane >> 4;  // K half-select

  const float* aRow = &sA[(wave * 16 + mrow) * DF];

  float bias[NTILES];
#pragma unroll
  for (int t = 0; t < NTILES; ++t) bias[t] = b[t * 16 + mrow];

  v8f acc[NTILES] = {};

  for (int k = 0; k < DF; k += 4) {
    // A fragment (16x4 f32): lanes 0-15 -> K = k,k+1 ; lanes 16-31 -> K = k+2,k+3
    v2f a;
    a.x = aRow[k + 2 * kh + 0];
    a.y = aRow[k + 2 * kh + 1];
#pragma unroll
    for (int t = 0; t < NTILES; ++t) {
      // B fragment (4x16 f32): VGPR v, lane-half kh -> K = k + v + 2*kh
      v2f bf;
      bf.x = sW[(k + 2 * kh + 0) * DF + t * 16 + mrow];
      bf.y = sW[(k + 2 * kh + 1) * DF + t * 16 + mrow];
      acc[t] = __builtin_amdgcn_wmma_f32_16x16x4_f32(
          /*neg_a=*/false, a, /*neg_b=*/false, bf,
          /*c_mod=*/(short)0, acc[t], /*reuse_a=*/false, /*reuse_b=*/false);
    }
  }

  // D tile layout: VGPR i -> row i + 8*kh, col = lane%16
#pragma unroll
  for (int t = 0; t < NTILES; ++t) {
#pragma unroll
    for (int i = 0; i < 8; ++i) {
      int r = rowBase + wave * 16 + i + kh * 8;
      if (r < N) out[(size_t)r * DF + t * 16 + mrow] = acc[t][i] + bias[t];
    }
  }
}

extern "C" void kernel_launch(void* const* d_in, const int* in_sizes, int n_in,
                              void* d_out, int out_size, void* d_ws, size_t ws_size,
                              hipStream_t stream) {
  const float* feat = (const float*)d_in[0];
  const int*   src  = (const int*)d_in[1];
  const int*   dst  = (const int*)d_in[2];
  const float* W    = (const float*)d_in[3];
  const float* b    = (const float*)d_in[4];
  float*       out  = (float*)d_out;

  const int N = in_sizes[0] / DF;   // 50000
  const int E = in_sizes[1];        // 800000

  unsigned int* mx  = (unsigned int*)d_ws;
  float* num = (float*)((char*)d_ws + (size_t)1 * N * DF * sizeof(float));
  float* den = (float*)((char*)d_ws + (size_t)2 * N * DF * sizeof(float));

  // 1) zero max/num/den (contiguous 3*N*DF words)
  const int totZero = 3 * N * DF;
  genconv_zero_ws<<<(totZero + 255) / 256, 256, 0, stream>>>(mx, totZero);

  // 2) segment max, 3) fused exp-sum passes over edges (one thread per 4 dims)
  const int totE = E * 24;
  const int egrid = (totE + 255) / 256;
  genconv_edge_max<<<egrid, 256, 0, stream>>>(feat, src, dst, mx, E);
  genconv_edge_sum<<<egrid, 256, 0, stream>>>(feat, src, dst, mx, num, den, E);

  // 4) residual + WMMA GEMM + bias
  const int ggrid = (N + TILE_ROWS - 1) / TILE_ROWS;
  genconv_fused_gemm<<<ggrid, GEMM_THREADS, 0, stream>>>(feat, num, den, W, b,
                                                         out, N);
}